// Temporal_voting_FC1_89833535963827
// MI455X (gfx1250) — compile-verified
//
#include <hip/hip_runtime.h>
#include <math.h>

typedef float v2f __attribute__((ext_vector_type(2)));
typedef float v4f __attribute__((ext_vector_type(4)));
typedef float v8f __attribute__((ext_vector_type(8)));

#define NUM_CLASSES 29
#define IN_CH 1152
#define B_STRIDE (2 * IN_CH + 4)   // 2308 floats; 2308 % 64 == 4 -> conflict-free
#define B_ROWS 16                  // row lh holds classes lh and lh+16 interleaved
#define WAVES_PER_WG 8
#define FRAMES_PER_WAVE 16
#define FRAMES_PER_WG (WAVES_PER_WG * FRAMES_PER_WAVE)   // 128

// ---------------------------------------------------------------------------
// Kernel 0: zero the global histogram in workspace
// ---------------------------------------------------------------------------
__global__ void tv_init(unsigned int* __restrict__ g_hist) {
    int i = threadIdx.x;
    if (i < 32) g_hist[i] = 0u;
}

// ---------------------------------------------------------------------------
// Kernel 1: WMMA GEMM (per-frame logits) + per-frame argmax + histogram
// ---------------------------------------------------------------------------
__global__ __launch_bounds__(WAVES_PER_WG * 32)
void tv_main(const float* __restrict__ x,
             const float* __restrict__ W,
             const float* __restrict__ bias,
             unsigned int* __restrict__ g_hist) {
    // Interleaved B: row lh, K-pair j -> { W[lh][2j], W[lh][2j+1], W[lh+16][2j], W[lh+16][2j+1] }
    __shared__ float        lds_w[B_ROWS * B_STRIDE];   // ~148 KB, 2 WGs / 320KB WGP
    __shared__ unsigned int lds_hist[32];

    const int tid = threadIdx.x;

    // ---- clear (zeroes phantom classes 29..31), then stage W interleaved ----
    for (int i = tid; i < B_ROWS * B_STRIDE; i += blockDim.x) lds_w[i] = 0.0f;
    if (tid < 32) lds_hist[tid] = 0u;
    __syncthreads();
    for (int i = tid; i < NUM_CLASSES * IN_CH; i += blockDim.x) {
        int n = i / IN_CH;                 // class
        int c = i - n * IN_CH;             // channel
        int row     = n & 15;
        int tileoff = (n >> 4) * 2;        // 0 for classes 0..15, 2 for 16..28
        lds_w[row * B_STRIDE + (c >> 1) * 4 + tileoff + (c & 1)] = W[i];
    }
    __syncthreads();

    const int wave = tid >> 5;          // 0..7
    const int lane = tid & 31;
    const int half = lane >> 4;         // K-pair selector for f32 A/B fragments
    const int lh   = lane & 15;         // M (frames) for A, N (classes) for B
    const int kbase = half * 2;

    const int tileBase = (blockIdx.x * WAVES_PER_WG + wave) * FRAMES_PER_WAVE;

    // A fragment source: lane holds row (tileBase+lh), K = k+kbase, k+kbase+1
    const float* aptr = x + (size_t)(tileBase + lh) * IN_CH + kbase;

    // B fragments: one b128 per K-step yields both class tiles
    // offset (floats) for K-step k: 2k  (pair j = k/2 + half baked into base)
    const float* bptr = &lds_w[lh * B_STRIDE + half * 4];

    v8f c0 = {};   // logits for classes 0..15, frames tileBase..tileBase+15
    v8f c1 = {};   // logits for classes 16..31 (29..31 are 0, masked at argmax)

    for (int k0 = 0; k0 < IN_CH; k0 += 64) {
        // keep the x stream flowing from HBM/L2 ahead of use
        __builtin_prefetch(aptr + k0 + 256, 0, 1);
#pragma unroll
        for (int k = k0; k < k0 + 64; k += 4) {
            v2f a = *(const v2f*)(aptr + k);
            v4f b = *(const v4f*)(bptr + 2 * k);   // ds_load_b128: b0 | b1
            v2f b0; b0.x = b.x; b0.y = b.y;
            v2f b1; b1.x = b.z; b1.y = b.w;
            // D = A(16x4) * B(4x16) + C ; f32 accumulate on the matrix pipe
            c0 = __builtin_amdgcn_wmma_f32_16x16x4_f32(
                    false, a, false, b0, (short)0, c0, false, false);
            c1 = __builtin_amdgcn_wmma_f32_16x16x4_f32(
                    false, a, false, b1, (short)0, c1, false, false);
        }
    }

    // ---- bias + per-frame argmax over 29 classes ----
    const int   n1     = 16 + lh;
    const bool  valid1 = (n1 < NUM_CLASSES);
    const float bias0  = bias[lh];
    const float bias1  = valid1 ? bias[n1] : 0.0f;

    // C layout: VGPR r, lanes 0-15 -> frame r, lanes 16-31 -> frame r+8; N = lane&15
#pragma unroll
    for (int r = 0; r < 8; ++r) {
        float v  = c0[r] + bias0;
        int   cl = lh;
        float v1 = valid1 ? (c1[r] + bias1) : -INFINITY;
        if (v1 > v) { v = v1; cl = n1; }          // tie -> lower class (tile0)

        // butterfly argmax across the 16-lane half (xor<16 stays in-half)
#pragma unroll
        for (int off = 1; off < 16; off <<= 1) {
            float ov = __shfl_xor(v,  off, 32);
            int   oc = __shfl_xor(cl, off, 32);
            if (ov > v || (ov == v && oc < cl)) { v = ov; cl = oc; }
        }
        // one vote per frame: lane 0 of each half (frames r and r+8)
        if (lh == 0) atomicAdd(&lds_hist[cl], 1u);
    }

    __syncthreads();
    if (tid < NUM_CLASSES) {
        unsigned int h = lds_hist[tid];
        if (h) atomicAdd(&g_hist[tid], h);
    }
}

// ---------------------------------------------------------------------------
// Kernel 2: argmax over 29 bins (lowest index on tie) -> one-hot [1,29]
// ---------------------------------------------------------------------------
__global__ void tv_final(const unsigned int* __restrict__ g_hist,
                         float* __restrict__ out) {
    int i = threadIdx.x;
    unsigned int best = g_hist[0];
    int winner = 0;
    for (int c = 1; c < NUM_CLASSES; ++c) {
        unsigned int h = g_hist[c];
        if (h > best) { best = h; winner = c; }
    }
    if (i < NUM_CLASSES) out[i] = (i == winner) ? 1.0f : 0.0f;
}

// ---------------------------------------------------------------------------
extern "C" void kernel_launch(void* const* d_in, const int* in_sizes, int n_in,
                              void* d_out, int out_size, void* d_ws, size_t ws_size,
                              hipStream_t stream) {
    const float* x  = (const float*)d_in[0];   // [1, T, 1152] f32
    const float* W  = (const float*)d_in[1];   // [29, 1152]   f32
    const float* b  = (const float*)d_in[2];   // [29]         f32
    float* out = (float*)d_out;                // [1, 29]      f32

    unsigned int* g_hist = (unsigned int*)d_ws;

    const int T = in_sizes[0] / IN_CH;         // 65536
    const int grid = (T + FRAMES_PER_WG - 1) / FRAMES_PER_WG;   // 512

    tv_init<<<1, 32, 0, stream>>>(g_hist);
    tv_main<<<grid, WAVES_PER_WG * 32, 0, stream>>>(x, W, b, g_hist);
    tv_final<<<1, 32, 0, stream>>>(g_hist, out);
}